// PostProcess_37649683317170
// MI455X (gfx1250) — compile-verified
//
#include <hip/hip_runtime.h>
#include <hip/hip_bf16.h>
#include <stdint.h>

#define Q        2048
#define NIMG     16
#define NTHREADS 1024
#define IOU_TH   0.5f
#define CONF_TH  0.7f

typedef __attribute__((ext_vector_type(2))) float v2f;
typedef __attribute__((ext_vector_type(8))) float v8f;

static __device__ __forceinline__ uint32_t ballot32(bool p) {
    return __builtin_amdgcn_ballot_w32(p);
}

static __device__ __forceinline__ float boxArea(const float4 b) {
    return fmaxf(b.z - b.x, 0.f) * fmaxf(b.w - b.y, 0.f);
}

static __device__ __forceinline__ float boxIoU(const float4 a, const float4 b) {
    float ix = fminf(a.z, b.z) - fmaxf(a.x, b.x);
    float iy = fminf(a.w, b.w) - fmaxf(a.y, b.y);
    float inter = fmaxf(ix, 0.f) * fmaxf(iy, 0.f);
    float uni = boxArea(a) + boxArea(b) - inter;
    return inter / fmaxf(uni, 1e-9f);
}

__global__ __launch_bounds__(NTHREADS)
void PostProcess_37649683317170_kernel(const float* __restrict__ logits,
                                       const float* __restrict__ pboxes,
                                       const int*   __restrict__ imgH,
                                       const int*   __restrict__ imgW,
                                       float*        __restrict__ outScores,
                                       int*          __restrict__ outBoxes,
                                       unsigned char* __restrict__ outKeep) {
    __shared__ unsigned long long sKey[Q];     // (~score_bits)<<32 | orig_idx
    __shared__ float4   sBox[Q];               // scaled xyxy, original order
    __shared__ float    sScore[Q];
    __shared__ uint32_t sSupp[Q / 32];         // suppression bits (sorted pos)
    __shared__ uint32_t sKeepBits[Q / 32];     // keep bits (sorted pos)
    __shared__ float4   sBatchBox[16];
    __shared__ float    sBatchArea[16];
    __shared__ uint32_t sKeep16;
    __shared__ int      sVcnt;

    const int tid  = threadIdx.x;
    const int lane = tid & 31;
    const int wave = tid >> 5;
    const int img  = blockIdx.x;

    if (tid == 0) sVcnt = 0;
    if (tid < Q / 32) { sSupp[tid] = 0u; sKeepBits[tid] = 0u; }
    __syncthreads();

    // ---- Phase 1: score / validity / box transform ----
    const float W = (float)imgW[0];
    const float H = (float)imgH[0];
    const float*  lg = logits + (size_t)img * Q * 2;
    const float4* pb = (const float4*)(pboxes + (size_t)img * Q * 4);
    __builtin_prefetch(lg + 2 * tid, 0, 0);          // global_prefetch_b8
    __builtin_prefetch((const float*)pb + 4 * tid, 0, 0);

    for (int q = tid; q < Q; q += NTHREADS) {
        float z0 = lg[2 * q + 0], z1 = lg[2 * q + 1];
        float score = 1.f / (1.f + __expf(-fabsf(z0 - z1)));   // softmax max of 2
        bool  valid = (z0 >= z1) & (score >= CONF_TH);         // argmax==0 & conf
        float4 c = pb[q];
        float4 b;
        b.x = (c.x - 0.5f * c.z) * W;
        b.y = (c.y - 0.5f * c.w) * H;
        b.z = (c.x + 0.5f * c.z) * W;
        b.w = (c.y + 0.5f * c.w) * H;
        sBox[q]   = b;
        sScore[q] = score;
        uint32_t k = valid ? ~__float_as_uint(score) : 0xFFFFFFFFu; // asc => score desc, invalid last
        sKey[q] = ((unsigned long long)k << 32) | (uint32_t)q;       // idx => stable ties
        if (valid) atomicAdd(&sVcnt, 1);
    }

    // ---- Phase 2: bitonic sort of 2048 u64 keys (1024 threads) ----
    for (unsigned k = 2; k <= Q; k <<= 1) {
        for (unsigned j = k >> 1; j > 0; j >>= 1) {
            __syncthreads();
            unsigned t = (unsigned)tid;
            unsigned i = ((t & ~(j - 1)) << 1) | (t & (j - 1));
            unsigned p = i | j;
            unsigned long long a = sKey[i], b = sKey[p];
            bool asc = ((i & k) == 0);
            if (asc ? (a > b) : (a < b)) { sKey[i] = b; sKey[p] = a; }
        }
    }
    __syncthreads();
    const int Vcnt = sVcnt;   // valid boxes occupy sorted prefix [0, Vcnt)

    // ---- Phase 3: greedy NMS, 16-wide batches ----
    const int numBatches = (Vcnt + 15) >> 4;
    for (int bb = 0; bb < numBatches; ++bb) {
        const int p0   = bb << 4;
        const int bcnt = min(16, Vcnt - p0);

        if (wave == 0) {   // intra-batch sequential resolution, ballot-driven
            const int  n    = lane & 15;
            const int  pos  = p0 + n;
            const int  posc = min(pos, Q - 1);
            const unsigned inb = (unsigned)(n < bcnt);
            uint32_t oidx = (uint32_t)sKey[posc];            // unconditional LDS read
            float4   bn   = sBox[oidx];
            unsigned gsb  = (sSupp[posc >> 5] >> (posc & 31)) & 1u;
            uint32_t gb   = ballot32((inb & gsb) != 0u);
            uint32_t blocked = (gb | (gb >> 16)) & 0xFFFFu;  // suppressed by earlier batches
            uint32_t keep16 = 0;
            for (int i = 0; i < bcnt; ++i) {
                if (!((blocked >> i) & 1u)) {                // uniform (scalar) branch
                    keep16 |= 1u << i;
                    uint32_t oi = (uint32_t)sKey[p0 + i];
                    float4 bi = sBox[oi];
                    unsigned pred = inb & (unsigned)(n > i)
                                        & (unsigned)(boxIoU(bi, bn) > IOU_TH);
                    uint32_t pbm = ballot32(pred != 0u);
                    blocked |= (pbm | (pbm >> 16)) & 0xFFFFu;
                }
            }
            if (lane < 16) {
                sBatchBox[lane]  = bn;
                sBatchArea[lane] = inb ? boxArea(bn) : 0.f;
            }
            if (lane == 0) {
                sKeepBits[p0 >> 5] |= keep16 << (p0 & 31);
                sKeep16 = keep16;
            }
        }
        __syncthreads();

        const uint32_t keep16 = sKeep16;
        const int tailStart = p0 + 16;
        const int nTail = Vcnt - tailStart;
        if (keep16 != 0u && nTail > 0) {
            const int tiles = (nTail + 15) >> 4;
            const int hi8   = (lane & 16) >> 1;          // 0 for lanes 0-15, 8 for 16-31
            const bool lo   = (lane < 16);
            const float flo = lo ? 1.f : 0.f;
            const float aLo = sBatchArea[lane & 15];     // unconditional LDS read
            for (int t = wave; t < tiles; t += 32) {     // wave-uniform; EXEC all-1s
                const int  jbase = tailStart + (t << 4);
                const int  j     = jbase + (lane & 15);
                const unsigned inR = (unsigned)(j < Vcnt);
                uint32_t oj = (uint32_t)sKey[min(j, Q - 1)];
                float4 bj = sBox[oj];
                float areaj = inR ? boxArea(bj) : 0.f;

                // WMMA outer-sum: D[m][n] = batchArea[m] + areaj[n]
                // A(16x4): col0 = areas, col1 = 1 ; B(4x16): row0 = 1, row1 = areas_j
                v2f A, Bv;
                A[0]  = aLo * flo;        // K=0 rows (lanes 16-31 carry K=2 -> 0)
                A[1]  = flo;              // K=1
                Bv[0] = flo;              // row K=0 = ones
                Bv[1] = areaj * flo;      // row K=1 = areas_j
                v8f C = {};
                C = __builtin_amdgcn_wmma_f32_16x16x4_f32(
                        false, A, false, Bv, (short)0, C, false, false);

                unsigned anyMask = 0u;
                #pragma unroll
                for (int r = 0; r < 8; ++r) {
                    const int m = r + hi8;
                    float4 bm = sBatchBox[m];             // unconditional ds_load_b128
                    float ix = fminf(bm.z, bj.z) - fmaxf(bm.x, bj.x);
                    float iy = fminf(bm.w, bj.w) - fmaxf(bm.y, bj.y);
                    float inter = fmaxf(ix, 0.f) * fmaxf(iy, 0.f);
                    float uni = C[r] - inter;             // area_m + area_j - inter
                    unsigned sup = (unsigned)(inter > IOU_TH * fmaxf(uni, 1e-9f));
                    anyMask |= ((keep16 >> m) & 1u) & sup;   // branchless gate
                }
                uint32_t abal = ballot32((anyMask & inR) != 0u);
                uint32_t bits = (abal | (abal >> 16)) & 0xFFFFu;
                if (lane == 0 && bits)
                    atomicOr(&sSupp[jbase >> 5], bits << (jbase & 31));
            }
        }
        __syncthreads();
    }

    // ---- Phase 4: scatter outputs (scores f32, boxes i32 trunc, keep u8) ----
    for (int s = tid; s < Q; s += NTHREADS) {
        unsigned long long kk = sKey[s];
        uint32_t oidx = (uint32_t)kk;
        bool kept = (sKeepBits[s >> 5] >> (s & 31)) & 1u;
        float4 b = sBox[oidx];
        float  sc = sScore[oidx];
        size_t g = (size_t)img * Q + oidx;
        outScores[g] = kept ? sc : 0.f;
        int4 ib;
        ib.x = kept ? (int)b.x : 0;
        ib.y = kept ? (int)b.y : 0;
        ib.z = kept ? (int)b.z : 0;
        ib.w = kept ? (int)b.w : 0;
        ((int4*)outBoxes)[g] = ib;
        outKeep[g] = kept ? (unsigned char)1 : (unsigned char)0;
    }
}

extern "C" void kernel_launch(void* const* d_in, const int* in_sizes, int n_in,
                              void* d_out, int out_size, void* d_ws, size_t ws_size,
                              hipStream_t stream) {
    const float* logits = (const float*)d_in[0];
    const float* boxes  = (const float*)d_in[1];
    const int*   imgH   = (const int*)d_in[2];
    const int*   imgW   = (const int*)d_in[3];

    // Output tuple concatenated flat: scores [B,Q] f32, boxes [B,Q,4] i32, keep [B,Q] u8
    float*         outScores = (float*)d_out;
    int*           outBoxes  = (int*)(outScores + (size_t)NIMG * Q);
    unsigned char* outKeep   = (unsigned char*)(outBoxes + (size_t)NIMG * Q * 4);

    PostProcess_37649683317170_kernel<<<dim3(NIMG), dim3(NTHREADS), 0, stream>>>(
        logits, boxes, imgH, imgW, outScores, outBoxes, outKeep);
}